// SAM_11716670783967
// MI455X (gfx1250) — compile-verified
//
#include <hip/hip_runtime.h>

// CDNA5 / gfx1250 fused 1x1-conv + BN + bilinear-pooling.
// - fp32 matrix pipe (v_wmma_f32_16x16x4_f32) for both GEMMs (exact precision)
// - GLOBAL_LOAD_ASYNC_TO_LDS_B128 double-buffered pipeline (ASYNCcnt) for all staging
// - linear LDS activation tiles: global chunk is a contiguous 25KB memcpy

typedef __attribute__((ext_vector_type(2))) float v2f;
typedef __attribute__((ext_vector_type(8))) float v8f;

#define C_     2048
#define K_     16
#define HW_    49
#define CHUNK  128                  // channels per LDS tile
#define NVEC_F 1568                 // CHUNK*HW_/4 float4s per F chunk
#define FPAD   16                   // zeroed tail pad (covers N/K over-reads)
#define FELEM  (CHUNK * HW_ + FPAD) // 6288 floats
#define WSTR   132                  // W row stride (132 % 64 == 4 -> conflict-free A reads)
#define WELEM  (K_ * WSTR)
#define PSTR   17

#define WMMA_F32(a, b, c) \
  __builtin_amdgcn_wmma_f32_16x16x4_f32(false, (a), false, (b), (short)0, (c), false, false)

// Raw LDS byte offset for async-to-LDS VDST operand (flat->group addrspacecast).
__device__ __forceinline__ unsigned lds_u32(const void* p) {
  return (unsigned)(unsigned long long)(__attribute__((address_space(3))) const char*)p;
}

__device__ __forceinline__ void async_b128(unsigned lds, unsigned voff, const float* sbase) {
  asm volatile("global_load_async_to_lds_b128 %0, %1, %2"
               :: "v"(lds), "v"(voff), "s"(sbase) : "memory");
}

// Exactly 17 async instructions per wave per call (13 F + 4 W) -> constant s_wait_asynccnt.
__device__ __forceinline__ void prefetch_chunk(const float* f_cc, const float* w_cc,
                                               unsigned ldsF, unsigned ldsW, int tid) {
  #pragma unroll
  for (int j = 0; j < 13; ++j) {
    int idx = tid + j * 128;
    if (idx > NVEC_F - 1) idx = NVEC_F - 1;   // clamp (duplicate copy) keeps count uniform
    unsigned voff = (unsigned)idx << 4;
    async_b128(ldsF + voff, voff, f_cc);
  }
  #pragma unroll
  for (int j = 0; j < 4; ++j) {
    int idx = tid + j * 128;                  // < 512
    int k = idx >> 5, jj = idx & 31;
    async_b128(ldsW + (unsigned)(k * WSTR + jj * 4) * 4u,
               (unsigned)(k * C_ + jj * 4) * 4u, w_cc);
  }
}

__global__ __launch_bounds__(128) void sam_bilinear_wmma(
    const float* __restrict__ featmap, const float* __restrict__ w16,
    const float* __restrict__ gamma, const float* __restrict__ beta,
    const float* __restrict__ mean, const float* __restrict__ var,
    float* __restrict__ out)
{
  __shared__ __align__(16) float sF[2][FELEM];   // linear activation chunks (double buffer)
  __shared__ __align__(16) float sW[2][WELEM];   // W chunks, row stride 132
  __shared__ float sPt[52 * PSTR];               // BN'd P transposed [hw][k], rows 49..51 zero
  __shared__ float sInv[16], sBias[16];

  const int b    = blockIdx.x;
  const int tid  = threadIdx.x;
  const int wave = tid >> 5;                     // 4 waves (wave32)
  const int lane = tid & 31;
  const int l16  = lane & 15;
  const int hi   = lane >> 4;                    // selects K pair {0,1} vs {2,3}

  if (tid < 16) {
    float inv  = gamma[tid] * rsqrtf(var[tid] + 1e-5f);
    sInv[tid]  = inv;
    sBias[tid] = beta[tid] - mean[tid] * inv;
    sF[0][CHUNK * HW_ + tid] = 0.0f;             // zero tail pads (finite over-reads)
    sF[1][CHUNK * HW_ + tid] = 0.0f;
  }

  const float* Fb = featmap + (size_t)b * (C_ * HW_);
  float* outB     = out + (size_t)b * (K_ * C_);
  const unsigned ldsF0 = lds_u32(&sF[0][0]), ldsF1 = lds_u32(&sF[1][0]);
  const unsigned ldsW0 = lds_u32(&sW[0][0]), ldsW1 = lds_u32(&sW[1][0]);

  prefetch_chunk(Fb, w16, ldsF0, ldsW0, tid);    // chunk 0 -> buf 0

  v8f accP = {};
  const int hwb = wave * 16;

  // 16 pipeline steps: i<8 = GEMM1 (P = W x F), i>=8 = GEMM2 (M = P x F^T); featmap
  // is streamed twice, second pass hits the 192MB L2 (whole featmap = 102MB resident).
  for (int i = 0; i < 16; ++i) {
    const int cur = i & 1;
    if (i < 15) {
      const int nc = (i + 1) & 7;
      prefetch_chunk(Fb + nc * (CHUNK * HW_), w16 + nc * CHUNK,
                     cur ? ldsF0 : ldsF1, cur ? ldsW0 : ldsW1, tid);
      asm volatile("s_wait_asynccnt 17" ::: "memory");  // chunk i complete (in-order)
    } else {
      asm volatile("s_wait_asynccnt 0" ::: "memory");
    }
    __syncthreads();   // all waves' async writes of chunk i visible

    const float* Fbuf = sF[cur];
    if (i < 8) {
      // ---- GEMM1: each wave owns one hw-tile (N=16) of P, K-loop over channels ----
      const float* Wbuf = sW[cur];
      #pragma unroll 4
      for (int kk = 0; kk < CHUNK; kk += 4) {
        v2f a, bm;
        a.x  = Wbuf[l16 * WSTR + kk + 2 * hi];          // pairs into ds_load_b64
        a.y  = Wbuf[l16 * WSTR + kk + 2 * hi + 1];
        bm.x = Fbuf[(kk + 2 * hi)     * HW_ + hwb + l16]; // coalesced rows
        bm.y = Fbuf[(kk + 2 * hi + 1) * HW_ + hwb + l16];
        accP = WMMA_F32(a, bm, accP);
      }
      if (i == 7) {
        // BN in-register; store P transposed; hw>=49 garbage columns -> exact zeros
        int hw = hwb + l16;
        if (hw < 52) {
          #pragma unroll
          for (int r = 0; r < 8; ++r) {
            int k = r + 8 * hi;                          // C/D rows r and r+8 per lane half
            float v = accP[r] * sInv[k] + sBias[k];
            sPt[hw * PSTR + k] = (hw < HW_) ? v : 0.0f;
          }
        }
      }
    } else {
      // ---- GEMM2: M[16, chunk] = Pt^T x F^T, K = hw padded to 52 (Pt rows 49..51 = 0) ----
      const int cc = (i & 7) * CHUNK;
      for (int t = wave; t < CHUNK / 16; t += 4) {       // wave-uniform, EXEC all-1s
        const int nb = t * 16;
        v8f acc = {};
        #pragma unroll
        for (int kk = 0; kk < 52; kk += 4) {
          v2f a, bm;
          a.x  = sPt[(kk + 2 * hi)     * PSTR + l16];
          a.y  = sPt[(kk + 2 * hi + 1) * PSTR + l16];
          bm.x = Fbuf[(nb + l16) * HW_ + kk + 2 * hi];   // stride-49 -> conflict-free
          bm.y = Fbuf[(nb + l16) * HW_ + kk + 2 * hi + 1];
          acc = WMMA_F32(a, bm, acc);
        }
        #pragma unroll
        for (int r = 0; r < 8; ++r) {
          int k = r + 8 * hi;
          outB[k * C_ + cc + nb + l16] = acc[r] * (1.0f / 49.0f);  // coalesced per row
        }
      }
    }
    __syncthreads();   // buf[cur] free before it is refilled at step i+1
  }
}

extern "C" void kernel_launch(void* const* d_in, const int* in_sizes, int n_in,
                              void* d_out, int out_size, void* d_ws, size_t ws_size,
                              hipStream_t stream) {
  const float* featmap = (const float*)d_in[0];
  const float* w16     = (const float*)d_in[1];
  const float* gamma   = (const float*)d_in[2];
  const float* beta    = (const float*)d_in[3];
  const float* mean    = (const float*)d_in[4];
  const float* var     = (const float*)d_in[5];
  float* outp          = (float*)d_out;

  const int B = in_sizes[0] / (C_ * HW_);   // 256
  sam_bilinear_wmma<<<dim3(B), dim3(128), 0, stream>>>(
      featmap, w16, gamma, beta, mean, var, outp);
}